// MyGCNLayer_58557584113802
// MI455X (gfx1250) — compile-verified
//
#include <hip/hip_runtime.h>
#include <hip/hip_bf16.h>
#include <stdint.h>

// GCN layer: out = D^-1 (A + I) D^-1 (x W^T), N=100000, C=128, E=1600000.
// Strategy (MI455X):
//  - edge aggregation dominates (~820MB random L2-resident gathers); build CSR by src
//    (counting sort) so the gather pass needs NO output atomics, one wave32 per node,
//    float4/lane coalesced 512B reads, self-loop + 1/d scale fused into the same pass.
//  - h = x@W.T via V_WMMA_F32_16X16X4_F32 (full f32 precision; GEMM is only 3.3 GFLOP).

typedef __attribute__((ext_vector_type(2))) float v2f;
typedef __attribute__((ext_vector_type(8))) float v8f;

#define C_DIM 128

// ---------------- K1: zero the two histograms ----------------
__global__ void gcn_init_counts(int* __restrict__ cnt_src, int* __restrict__ cnt_dst, int n) {
    int i = blockIdx.x * blockDim.x + threadIdx.x;
    if (i < n) { cnt_src[i] = 0; cnt_dst[i] = 0; }
}

// ---------------- K2: per-edge histograms (src for CSR, dst for degree) ----------------
__global__ void gcn_count_edges(const long long* __restrict__ ei,
                                int* __restrict__ cnt_src, int* __restrict__ cnt_dst, int E) {
    int e = blockIdx.x * blockDim.x + threadIdx.x;
    if (e < E) {
        int s = (int)ei[e];
        int d = (int)ei[(size_t)E + e];
        atomicAdd(&cnt_src[s], 1);
        atomicAdd(&cnt_dst[d], 1);
    }
}

// ---------------- K3: invdeg[i] = 1 / (indeg + 1 self loop) ----------------
__global__ void gcn_invdeg(const int* __restrict__ cnt_dst, float* __restrict__ invdeg, int n) {
    int i = blockIdx.x * blockDim.x + threadIdx.x;
    if (i < n) invdeg[i] = 1.0f / (float)(cnt_dst[i] + 1);
}

// ---------------- K4: single-block exclusive scan -> row_ptr (+cursor copy) ----------------
__global__ void gcn_scan(const int* __restrict__ cnt, int* __restrict__ row_ptr,
                         int* __restrict__ cursor, int n) {
    __shared__ int sdata[1024];
    __shared__ int s_run;
    if (threadIdx.x == 0) s_run = 0;
    __syncthreads();
    for (int base = 0; base < n; base += 1024) {
        int i = base + (int)threadIdx.x;
        int v = (i < n) ? cnt[i] : 0;
        sdata[threadIdx.x] = v;
        __syncthreads();
        #pragma unroll
        for (int off = 1; off < 1024; off <<= 1) {
            int t = (threadIdx.x >= (unsigned)off) ? sdata[threadIdx.x - off] : 0;
            __syncthreads();
            sdata[threadIdx.x] += t;
            __syncthreads();
        }
        int incl = sdata[threadIdx.x];
        int run  = s_run;                 // read before update
        if (i < n) {
            int rp = run + (incl - v);    // exclusive
            row_ptr[i] = rp;
            cursor[i]  = rp;
        }
        __syncthreads();
        if (threadIdx.x == 1023) s_run = run + sdata[1023];
        __syncthreads();
    }
    if (threadIdx.x == 0) row_ptr[n] = s_run;
}

// ---------------- K5: place edges into CSR slots ----------------
__global__ void gcn_place_edges(const long long* __restrict__ ei,
                                int* __restrict__ cursor, int* __restrict__ col, int E) {
    int e = blockIdx.x * blockDim.x + threadIdx.x;
    if (e < E) {
        int s = (int)ei[e];
        int d = (int)ei[(size_t)E + e];
        int slot = atomicAdd(&cursor[s], 1);
        col[slot] = d;
    }
}

// ---------------- K6: hn = (x @ W^T) * invdeg[row], WMMA f32 16x16x4 ----------------
// One wave32 per 16-row strip, all 128 output cols (8 x v8f accumulators).
// f32 A-frag (16x4): lanes 0-15 hold M=0..15 K={k,k+1}, lanes 16-31 hold K={k+2,k+3}.
// f32 B-frag (4x16): symmetric; B[k,o] = W[o,k] so both frags load v2f at (row,kk).
__global__ void __launch_bounds__(256)
gcn_gemm_scale(const float* __restrict__ x, const float* __restrict__ W,
               const float* __restrict__ invdeg, float* __restrict__ hn, int nTiles) {
    int tile = blockIdx.x * 8 + (threadIdx.x >> 5);
    if (tile >= nTiles) return;                  // uniform per wave: EXEC stays all-1s
    int lane = threadIdx.x & 31;
    int half = lane >> 4;                        // 0 or 1
    int m16  = lane & 15;
    int row_base = tile * 16;

    const float* xrow = x + (size_t)(row_base + m16) * C_DIM;

    v8f acc[8] = {};
    for (int k = 0; k < C_DIM; k += 4) {
        int kk = k + half * 2;
        v2f a = *(const v2f*)(xrow + kk);
        #pragma unroll
        for (int c = 0; c < 8; ++c) {
            const float* wrow = W + (size_t)(c * 16 + m16) * C_DIM;
            v2f b = *(const v2f*)(wrow + kk);
            acc[c] = __builtin_amdgcn_wmma_f32_16x16x4_f32(
                false, a, false, b, (short)0, acc[c], false, false);
        }
    }

    // C/D layout: vgpr r -> row = r + 8*half, col = m16 (+16c)
    float idg[8];
    #pragma unroll
    for (int r = 0; r < 8; ++r) idg[r] = invdeg[row_base + 8 * half + r];
    #pragma unroll
    for (int c = 0; c < 8; ++c) {
        #pragma unroll
        for (int r = 0; r < 8; ++r) {
            size_t off = (size_t)(row_base + 8 * half + r) * C_DIM + c * 16 + m16;
            hn[off] = acc[c][r] * idg[r];
        }
    }
}

// ---------------- K7: gather, one wave32 per node, no atomics ----------------
// out[i] = invdeg[i] * ( hn[i] + sum_{j in CSR row i} hn[col[j]] )
__global__ void __launch_bounds__(256)
gcn_gather(const int* __restrict__ row_ptr, const int* __restrict__ col,
           const float* __restrict__ hn, const float* __restrict__ invdeg,
           float* __restrict__ out, int n) {
    int node = (int)((blockIdx.x * blockDim.x + threadIdx.x) >> 5);
    int lane = threadIdx.x & 31;
    if (node >= n) return;

    const float4* hn4 = (const float4*)hn;
    float4 acc = hn4[(size_t)node * 32 + lane];        // self-loop term hn[i]
    int beg = row_ptr[node];
    int end = row_ptr[node + 1];
    for (int j = beg; j < end; ++j) {
        int d = col[j];
        if (j + 1 < end)                                // hide L2 latency of next row
            __builtin_prefetch(hn + (size_t)col[j + 1] * C_DIM + lane * 4, 0, 0);
        float4 v = hn4[(size_t)d * 32 + lane];
        acc.x += v.x; acc.y += v.y; acc.z += v.z; acc.w += v.w;
    }
    float s = invdeg[node];
    float4 r; r.x = acc.x * s; r.y = acc.y * s; r.z = acc.z * s; r.w = acc.w * s;
    ((float4*)out)[(size_t)node * 32 + lane] = r;
}

static inline size_t align16(size_t v) { return (v + 15) & ~(size_t)15; }

extern "C" void kernel_launch(void* const* d_in, const int* in_sizes, int n_in,
                              void* d_out, int out_size, void* d_ws, size_t ws_size,
                              hipStream_t stream) {
    const float*     x  = (const float*)d_in[0];
    const long long* ei = (const long long*)d_in[1];   // int64 [2, E], src row then dst row
    const float*     W  = (const float*)d_in[2];       // [C_out, C_in]
    float*           out = (float*)d_out;

    const int N = in_sizes[0] / C_DIM;   // 100000
    const int E = in_sizes[1] / 2;       // 1600000

    // Workspace carve-up (~60 MB total)
    char* p = (char*)d_ws;
    float* invdeg  = (float*)p; p += align16((size_t)N * 4);
    int*   cnt_src = (int*)p;   p += align16((size_t)N * 4);
    int*   cnt_dst = (int*)p;   p += align16((size_t)N * 4);
    int*   row_ptr = (int*)p;   p += align16((size_t)(N + 1) * 4);
    int*   cursor  = (int*)p;   p += align16((size_t)N * 4);
    int*   col     = (int*)p;   p += align16((size_t)E * 4);
    float* hn      = (float*)p; // N * 128 floats

    int nbN = (N + 255) / 256;
    int nbE = (E + 255) / 256;

    gcn_init_counts<<<nbN, 256, 0, stream>>>(cnt_src, cnt_dst, N);
    gcn_count_edges<<<nbE, 256, 0, stream>>>(ei, cnt_src, cnt_dst, E);
    gcn_invdeg<<<nbN, 256, 0, stream>>>(cnt_dst, invdeg, N);
    gcn_scan<<<1, 1024, 0, stream>>>(cnt_src, row_ptr, cursor, N);
    gcn_place_edges<<<nbE, 256, 0, stream>>>(ei, cursor, col, E);

    int nTiles = (N + 15) / 16;          // N = 100000 -> 6250 exact
    int nbG = (nTiles + 7) / 8;
    gcn_gemm_scale<<<nbG, 256, 0, stream>>>(x, W, invdeg, hn, nTiles);

    long long gthreads = (long long)N * 32;
    int nbA = (int)((gthreads + 255) / 256);
    gcn_gather<<<nbA, 256, 0, stream>>>(row_ptr, col, hn, invdeg, out, N);
}